// HANLayer_59167469470139
// MI455X (gfx1250) — compile-verified
//
#include <hip/hip_runtime.h>
#include <math.h>

#define N_NODES 100000
#define N_EDGES 1600000
#define N_MP    3
#define D_IN    32
#define D_OUT   32
#define HID     128
#define NEG_SLOPE 0.2f

// N_NODES % 16 == 0 -> 6250 tiles per metapath, no row guards needed.
#define TILES_PER_MP (N_NODES / 16)
// 6250 = 5 * 1250: exact division -> constant trip count, EXEC stays all-ones
// through every WMMA (no tail, no exec masking).
#define TILES_PER_WAVE 5
#define WAVES_PER_MP   (TILES_PER_MP / TILES_PER_WAVE)   // 1250

typedef __attribute__((ext_vector_type(2))) float v2f;
typedef __attribute__((ext_vector_type(8))) float v8f;

// ---------------------------------------------------------------------------
// Kernel 1: z[mp] = h @ W[mp] via V_WMMA_F32_16X16X4_F32 (exact fp32 MACC).
// One wave: loads the 16 B fragments of W[mp] once (8 K-steps x 2 col halves,
// held in VGPRs), then sweeps 5 16-row node tiles -> 80 WMMA per wave.
// ---------------------------------------------------------------------------
__global__ void gemm_z_kernel(const float* __restrict__ h,   // [N, 32]
                              const float* __restrict__ W,   // [MP, 32, 32]
                              float* __restrict__ z)         // [MP, N, 32]
{
    const int wave = (blockIdx.x * blockDim.x + threadIdx.x) >> 5;
    const int lane = threadIdx.x & 31;
    const int mp   = wave / WAVES_PER_MP;                    // wave-uniform
    if (mp >= N_MP) return;
    const int tbase = (wave - mp * WAVES_PER_MP) * TILES_PER_WAVE;

    const int mrow = lane & 15;              // A row owned by this lane
    const int kgrp = lane >> 4;              // 0 -> K,K+1 ; 1 -> K+2,K+3
    const int ncol = lane & 15;              // B/C column owned by this lane
    const float* __restrict__ Wm = W + (size_t)mp * D_IN * D_OUT;

    // B fragments for the whole K=32 sweep, both 16-column halves.
    v2f bf0[8], bf1[8];
#pragma unroll
    for (int ks = 0; ks < 8; ++ks) {
        const int ka = ks * 4 + 2 * kgrp;
        bf0[ks].x = Wm[(ka + 0) * D_OUT + ncol];
        bf0[ks].y = Wm[(ka + 1) * D_OUT + ncol];
        bf1[ks].x = Wm[(ka + 0) * D_OUT + ncol + 16];
        bf1[ks].y = Wm[(ka + 1) * D_OUT + ncol + 16];
    }

#pragma unroll
    for (int t = 0; t < TILES_PER_WAVE; ++t) {               // constant trip
        const int tile = tbase + t;
        const float2* __restrict__ hrow =
            (const float2*)(h + (size_t)(tile * 16 + mrow) * D_IN);
        v2f af[8];
#pragma unroll
        for (int ks = 0; ks < 8; ++ks) {                     // float2 A loads
            const float2 v = hrow[ks * 2 + kgrp];
            af[ks].x = v.x;
            af[ks].y = v.y;
        }
        v8f c0 = {}, c1 = {};
#pragma unroll
        for (int ks = 0; ks < 8; ++ks) {
            c0 = __builtin_amdgcn_wmma_f32_16x16x4_f32(false, af[ks], false, bf0[ks],
                                                       (short)0, c0, false, false);
            c1 = __builtin_amdgcn_wmma_f32_16x16x4_f32(false, af[ks], false, bf1[ks],
                                                       (short)0, c1, false, false);
        }
        // C/D layout: VGPR r -> row (r + 8*kgrp), col (lane&15)
        float* __restrict__ zt =
            z + ((size_t)mp * N_NODES + tile * 16 + 8 * kgrp) * D_OUT;
#pragma unroll
        for (int r = 0; r < 8; ++r) {
            zt[r * D_OUT + ncol]      = c0[r];
            zt[r * D_OUT + ncol + 16] = c1[r];
        }
    }
}

// ---------------------------------------------------------------------------
// Kernel 2: per (mp,node): el/er scores + init emax/denom/agg/wsum.
// ---------------------------------------------------------------------------
__global__ void node_init_kernel(const float* __restrict__ z,
                                 const float* __restrict__ attn_l,  // [MP,32]
                                 const float* __restrict__ attn_r,  // [MP,32]
                                 float* __restrict__ el,
                                 float* __restrict__ er,
                                 float* __restrict__ emax,
                                 float* __restrict__ denom,
                                 float* __restrict__ agg,
                                 float* __restrict__ wsum)
{
    const int idx = blockIdx.x * blockDim.x + threadIdx.x;   // over MP*N
    if (idx < N_MP) wsum[idx] = 0.0f;
    if (idx >= N_MP * N_NODES) return;
    const int mp = idx / N_NODES;
    const float4* __restrict__ zr = (const float4*)(z + (size_t)idx * D_OUT);
    const float4* __restrict__ al = (const float4*)(attn_l + mp * D_OUT);
    const float4* __restrict__ ar = (const float4*)(attn_r + mp * D_OUT);
    float sl = 0.0f, sr = 0.0f;
#pragma unroll
    for (int q = 0; q < D_OUT / 4; ++q) {
        const float4 v = zr[q];
        const float4 a = al[q];
        const float4 b = ar[q];
        sl += v.x * a.x + v.y * a.y + v.z * a.z + v.w * a.w;
        sr += v.x * b.x + v.y * b.y + v.z * b.z + v.w * b.w;
    }
    el[idx]    = sl;
    er[idx]    = sr;
    emax[idx]  = -INFINITY;
    denom[idx] = 0.0f;
    float4* __restrict__ ag = (float4*)(agg + (size_t)idx * D_OUT);
    const float4 zero = {0.0f, 0.0f, 0.0f, 0.0f};
#pragma unroll
    for (int q = 0; q < D_OUT / 4; ++q) ag[q] = zero;
}

// IEEE-order float atomic max via int/uint monotonicity.
__device__ __forceinline__ void atomicMaxFloat(float* addr, float v) {
    if (v >= 0.0f) atomicMax((int*)addr, __float_as_int(v));
    else           atomicMin((unsigned int*)addr, __float_as_uint(v));
}

// ---------------------------------------------------------------------------
// Kernel 3: edge logits + segment max over dst.
// ---------------------------------------------------------------------------
__global__ void edge_max_kernel(const int* __restrict__ src,   // [MP,E]
                                const int* __restrict__ dst,   // [MP,E]
                                const float* __restrict__ el,
                                const float* __restrict__ er,
                                float* __restrict__ ebuf,      // [MP,E]
                                float* __restrict__ emax)
{
    const int idx = blockIdx.x * blockDim.x + threadIdx.x;     // over MP*E
    if (idx >= N_MP * N_EDGES) return;
    const int mp = idx / N_EDGES;
    const int s  = src[idx];
    const int d  = dst[idx];
    float x = el[mp * N_NODES + s] + er[mp * N_NODES + d];
    x = (x > 0.0f) ? x : NEG_SLOPE * x;                        // leaky relu
    ebuf[idx] = x;
    atomicMaxFloat(&emax[mp * N_NODES + d], x);
}

// ---------------------------------------------------------------------------
// Kernel 4: exp(e - emax[dst]) + segment sum over dst.
// ---------------------------------------------------------------------------
__global__ void edge_exp_kernel(const int* __restrict__ dst,
                                const float* __restrict__ emax,
                                float* __restrict__ ebuf,
                                float* __restrict__ denom)
{
    const int idx = blockIdx.x * blockDim.x + threadIdx.x;     // over MP*E
    if (idx >= N_MP * N_EDGES) return;
    const int mp = idx / N_EDGES;
    const int d  = dst[idx];
    const float ee = __expf(ebuf[idx] - emax[mp * N_NODES + d]);
    ebuf[idx] = ee;
    atomicAdd(&denom[mp * N_NODES + d], ee);
}

// ---------------------------------------------------------------------------
// Kernel 5: agg[dst] += alpha * z[src].  One wave per edge, lane = channel.
// Edge index derived from blockIdx + readfirstlane(warp id) so all per-edge
// scalars live on the SALU path; the z gather / agg atomics become
// SGPR-base + lane-offset accesses (fully coalesced 128B per wave).
// ---------------------------------------------------------------------------
__global__ void edge_agg_kernel(const int* __restrict__ src,
                                const int* __restrict__ dst,
                                const float* __restrict__ ebuf,
                                const float* __restrict__ denom,
                                const float* __restrict__ z,
                                float* __restrict__ agg)
{
    const int lane = threadIdx.x & 31;
    const int widx = __builtin_amdgcn_readfirstlane(threadIdx.x >> 5);
    const long long eidx = (long long)blockIdx.x * (blockDim.x >> 5) + widx;
    if (eidx >= (long long)N_MP * N_EDGES) return;
    const int mp = (int)(eidx / N_EDGES);
    const int s  = src[eidx];
    const int d  = dst[eidx];
    const float dn    = denom[mp * N_NODES + d];
    const float alpha = ebuf[eidx] / ((dn == 0.0f) ? 1.0f : dn);
    const float v = alpha * z[((size_t)mp * N_NODES + s) * D_OUT + lane];
    atomicAdd(&agg[((size_t)mp * N_NODES + d) * D_OUT + lane], v);
}

// ---------------------------------------------------------------------------
// Kernel 6: fused ELU (in place) + semantic projection via WMMA:
//   proj(n) = sum_j tanh((zfin @ W1)[n,j] + b1[j]) * w2[j]
// Only sum_n proj(n) is needed -> after each 16x16 WMMA tile, apply
// tanh/scale elementwise and reduce the whole tile to one scalar.
// W1 (16 KB) staged in LDS per block. One wave per (mp, 16-node tile),
// 64 WMMA per wave.
// ---------------------------------------------------------------------------
__global__ void finalize_kernel(float* __restrict__ agg,       // [MP,N,32] in: agg, out: elu(agg)
                                const float* __restrict__ W1,  // [32,128]
                                const float* __restrict__ b1,  // [128]
                                const float* __restrict__ w2,  // [128]
                                float* __restrict__ wsum)      // [MP]
{
    __shared__ float sW1[D_OUT * HID];                         // 16 KB
    for (int i = threadIdx.x; i < D_OUT * HID; i += blockDim.x)
        sW1[i] = W1[i];
    __syncthreads();

    const int wave = blockIdx.x * (blockDim.x >> 5) + (threadIdx.x >> 5);
    const int lane = threadIdx.x & 31;
    const int mp   = wave / TILES_PER_MP;                      // wave-uniform
    if (mp >= N_MP) return;
    const int tile = wave - mp * TILES_PER_MP;

    const int mrow = lane & 15;
    const int kgrp = lane >> 4;
    const int ncol = lane & 15;

    // ELU in place + build A fragments of zfin (f32 16x4 layout).
    float2* __restrict__ zrow =
        (float2*)(agg + ((size_t)mp * N_NODES + tile * 16 + mrow) * D_OUT);
    v2f af[8];
#pragma unroll
    for (int ks = 0; ks < 8; ++ks) {
        float2 v = zrow[ks * 2 + kgrp];
        v.x = (v.x > 0.0f) ? v.x : expm1f(v.x);
        v.y = (v.y > 0.0f) ? v.y : expm1f(v.y);
        zrow[ks * 2 + kgrp] = v;                               // disjoint per lane
        af[ks].x = v.x;
        af[ks].y = v.y;
    }

    float partial = 0.0f;
#pragma unroll
    for (int jt = 0; jt < HID / 16; ++jt) {                    // 8 column tiles
        const int j = jt * 16 + ncol;
        v8f acc = {};
#pragma unroll
        for (int ks = 0; ks < 8; ++ks) {
            const int ka = ks * 4 + 2 * kgrp;
            v2f b;
            b.x = sW1[(ka + 0) * HID + j];
            b.y = sW1[(ka + 1) * HID + j];
            acc = __builtin_amdgcn_wmma_f32_16x16x4_f32(false, af[ks], false, b,
                                                        (short)0, acc, false, false);
        }
        const float b1j = b1[j];
        const float w2j = w2[j];
#pragma unroll
        for (int r = 0; r < 8; ++r)
            partial += tanhf(acc[r] + b1j) * w2j;
    }
    // Butterfly reduce over the wave, one atomic per wave.
#pragma unroll
    for (int off = 16; off > 0; off >>= 1)
        partial += __shfl_xor(partial, off, 32);
    if (lane == 0) atomicAdd(&wsum[mp], partial);
}

// ---------------------------------------------------------------------------
// Kernel 7: beta = softmax(wsum / N)
// ---------------------------------------------------------------------------
__global__ void beta_kernel(const float* __restrict__ wsum, float* __restrict__ beta)
{
    if (threadIdx.x == 0 && blockIdx.x == 0) {
        const float w0 = wsum[0] / (float)N_NODES;
        const float w1 = wsum[1] / (float)N_NODES;
        const float w2 = wsum[2] / (float)N_NODES;
        const float m  = fmaxf(w0, fmaxf(w1, w2));
        const float e0 = __expf(w0 - m);
        const float e1 = __expf(w1 - m);
        const float e2 = __expf(w2 - m);
        const float inv = 1.0f / (e0 + e1 + e2);
        beta[0] = e0 * inv;
        beta[1] = e1 * inv;
        beta[2] = e2 * inv;
    }
}

// ---------------------------------------------------------------------------
// Kernel 8: out[n,d] = sum_mp beta[mp] * zfin[mp,n,d]   (float4 vectorized)
// ---------------------------------------------------------------------------
__global__ void output_kernel(const float* __restrict__ agg,   // [MP,N,32] (elu'd)
                              const float* __restrict__ beta,
                              float* __restrict__ out)         // [N,32]
{
    const int q = blockIdx.x * blockDim.x + threadIdx.x;       // over N*32/4
    if (q >= N_NODES * D_OUT / 4) return;
    const size_t stride4 = (size_t)N_NODES * D_OUT / 4;
    const float4* __restrict__ a4 = (const float4*)agg;
    const float b0 = beta[0], b1 = beta[1], b2 = beta[2];
    const float4 v0 = a4[q];
    const float4 v1 = a4[stride4 + q];
    const float4 v2 = a4[2 * stride4 + q];
    float4 o;
    o.x = b0 * v0.x + b1 * v1.x + b2 * v2.x;
    o.y = b0 * v0.y + b1 * v1.y + b2 * v2.y;
    o.z = b0 * v0.z + b1 * v1.z + b2 * v2.z;
    o.w = b0 * v0.w + b1 * v1.w + b2 * v2.w;
    ((float4*)out)[q] = o;
}

// ---------------------------------------------------------------------------
extern "C" void kernel_launch(void* const* d_in, const int* in_sizes, int n_in,
                              void* d_out, int out_size, void* d_ws, size_t ws_size,
                              hipStream_t stream)
{
    const float* h      = (const float*)d_in[0];   // [N,32]
    const int*   src    = (const int*)  d_in[1];   // [MP,E]
    const int*   dst    = (const int*)  d_in[2];   // [MP,E]
    const float* W      = (const float*)d_in[3];   // [MP,32,32]
    const float* attn_l = (const float*)d_in[4];   // [MP,32]
    const float* attn_r = (const float*)d_in[5];   // [MP,32]
    const float* W1     = (const float*)d_in[6];   // [32,128]
    const float* b1     = (const float*)d_in[7];   // [128]
    const float* w2     = (const float*)d_in[8];   // [128]
    float*       out    = (float*)d_out;           // [N,32]

    // Workspace layout (floats)
    float* ws = (float*)d_ws;
    const size_t znd = (size_t)N_MP * N_NODES * D_OUT;   // 9.6M
    const size_t mn  = (size_t)N_MP * N_NODES;           // 300K
    const size_t me  = (size_t)N_MP * N_EDGES;           // 4.8M
    float* z     = ws;            ws += znd;
    float* agg   = ws;            ws += znd;
    float* el    = ws;            ws += mn;
    float* er    = ws;            ws += mn;
    float* emax  = ws;            ws += mn;
    float* denom = ws;            ws += mn;
    float* ebuf  = ws;            ws += me;
    float* wsum  = ws;            ws += 8;
    float* beta  = ws;            ws += 8;

    // 1) feature GEMM (WMMA), 5 tiles/wave, no tail
    const int gemm_waves  = N_MP * WAVES_PER_MP;               // 3750
    const int gemm_blocks = (gemm_waves * 32 + 255) / 256;
    gemm_z_kernel<<<gemm_blocks, 256, 0, stream>>>(h, W, z);

    // 2) node scores + init
    const int node_blocks = (N_MP * N_NODES + 255) / 256;
    node_init_kernel<<<node_blocks, 256, 0, stream>>>(z, attn_l, attn_r,
                                                      el, er, emax, denom, agg, wsum);

    // 3-4) edge softmax passes
    const int edge_blocks = (N_MP * N_EDGES + 255) / 256;
    edge_max_kernel<<<edge_blocks, 256, 0, stream>>>(src, dst, el, er, ebuf, emax);
    edge_exp_kernel<<<edge_blocks, 256, 0, stream>>>(dst, emax, ebuf, denom);

    // 5) weighted aggregation (one wave per edge, lane = channel)
    const long long agg_waves = (long long)N_MP * N_EDGES;
    const int agg_blocks = (int)((agg_waves + 7) / 8);         // 8 waves/block
    edge_agg_kernel<<<agg_blocks, 256, 0, stream>>>(src, dst, ebuf, denom, z, agg);

    // 6) ELU + semantic projection (WMMA + LDS)
    const int fin_waves  = N_MP * TILES_PER_MP;
    const int fin_blocks = (fin_waves + 7) / 8;                // 8 waves/block
    finalize_kernel<<<fin_blocks, 256, 0, stream>>>(agg, W1, b1, w2, wsum);

    // 7) semantic softmax
    beta_kernel<<<1, 32, 0, stream>>>(wsum, beta);

    // 8) weighted combine
    const int out_blocks = (N_NODES * D_OUT / 4 + 255) / 256;
    output_kernel<<<out_blocks, 256, 0, stream>>>(agg, beta, out);
}